// GraphSAGE_87892210745356
// MI455X (gfx1250) — compile-verified
//
#include <hip/hip_runtime.h>

typedef __attribute__((ext_vector_type(2))) float v2f;
typedef __attribute__((ext_vector_type(8))) float v8f;

#define NN 100000
#define EE 1600000
#define DD 128
#define HH 128
#define CC 64

// ---------------------------------------------------------------------------
__global__ void zero_f32(float* __restrict__ p, size_t n) {
    size_t i = (size_t)blockIdx.x * blockDim.x + threadIdx.x;
    size_t stride = (size_t)gridDim.x * blockDim.x;
    for (; i < n; i += stride) p[i] = 0.0f;
}

__global__ void count_deg(const int* __restrict__ dst, float* __restrict__ deg, int ne) {
    int i = blockIdx.x * blockDim.x + threadIdx.x;
    if (i < ne) {
        __hip_atomic_fetch_add(&deg[dst[i]], 1.0f, __ATOMIC_RELAXED, __HIP_MEMORY_SCOPE_AGENT);
    }
}

__global__ void finalize_invdeg(float* __restrict__ deg, int n) {
    int i = blockIdx.x * blockDim.x + threadIdx.x;
    if (i < n) deg[i] = 1.0f / fmaxf(deg[i], 1.0f);
}

// ---------------------------------------------------------------------------
// edge scatter: one wave32 per edge, float4 gather per lane, f32 atomics into
// the L2-resident msg buffer.
__global__ void scatter_edges(const float* __restrict__ h,
                              const int* __restrict__ src,
                              const int* __restrict__ dst,
                              float* __restrict__ msg,
                              int ne) {
    int gid  = blockIdx.x * blockDim.x + threadIdx.x;
    int lane = gid & 31;
    int e    = gid >> 5;
    if (e >= ne) return;
    int s = __builtin_amdgcn_readfirstlane(src[e]);
    int d = __builtin_amdgcn_readfirstlane(dst[e]);
    const float4* hp = (const float4*)(h + (size_t)s * HH);
    float4 v = hp[lane];
    float* mp = msg + (size_t)d * HH + lane * 4;
    __hip_atomic_fetch_add(mp + 0, v.x, __ATOMIC_RELAXED, __HIP_MEMORY_SCOPE_AGENT);
    __hip_atomic_fetch_add(mp + 1, v.y, __ATOMIC_RELAXED, __HIP_MEMORY_SCOPE_AGENT);
    __hip_atomic_fetch_add(mp + 2, v.z, __ATOMIC_RELAXED, __HIP_MEMORY_SCOPE_AGENT);
    __hip_atomic_fetch_add(mp + 3, v.w, __ATOMIC_RELAXED, __HIP_MEMORY_SCOPE_AGENT);
}

// ---------------------------------------------------------------------------
__device__ __forceinline__ void async_ld16(unsigned lds_addr, const void* gaddr) {
    // GLOBAL_LOAD_ASYNC_TO_LDS_B128 vdst(LDS byte addr), vaddr(64b global), off
    asm volatile("global_load_async_to_lds_b128 %0, %1, off"
                 :: "v"(lds_addr), "v"(gaddr) : "memory");
}

// Fused SAGE GEMM: Out = act( A@W1 + (M*invdeg)@W2 + b )
// Block = 128 threads = 4 waves; each wave owns one 16-col tile and sweeps
// RT=5 row subtiles (80 rows/block). B fragments are register-resident;
// A/M stripes are staged to LDS once per block via async DMA.
template <int DOUT, bool HASN, bool RELU>
__global__ __launch_bounds__(128) void sage_gemm(
    const float* __restrict__ A,      // [N, 128]
    const float* __restrict__ M,      // [N, 128] (ignored unless HASN)
    const float* __restrict__ invdeg, // [N]
    const float* __restrict__ W1,     // [128, DOUT]
    const float* __restrict__ W2,     // [128, DOUT] (ignored unless HASN)
    const float* __restrict__ bias,   // [DOUT]
    float* __restrict__ Out)          // [N, DOUT]
{
    constexpr int DIN  = 128;
    constexpr int RT   = 5;     // 16-row subtiles per block
    constexpr int LDSW = 132;   // padded LDS row stride (floats) -> conflict-free b64 reads
    __shared__ float sA[RT * 16 * LDSW];
    __shared__ float sM[HASN ? RT * 16 * LDSW : 1];

    const int tid      = threadIdx.x;
    const int lane     = tid & 31;
    const int wave     = tid >> 5;
    const int row0     = blockIdx.x * (RT * 16);
    const int col_tile = blockIdx.y * 4 + wave;

    // ---- stage A (and M) stripe into LDS with async-to-LDS DMA
    {
        constexpr int NCHUNK = RT * 16 * (DIN / 4);  // 16-byte chunks = 2560
        #pragma unroll 4
        for (int c = tid; c < NCHUNK; c += 128) {
            const int r  = c >> 5;       // 32 chunks per 512B row
            const int cc = c & 31;
            unsigned lds = (unsigned)(uintptr_t)&sA[r * LDSW + cc * 4];
            async_ld16(lds, (const void*)(A + (size_t)(row0 + r) * DIN + cc * 4));
            if (HASN) {
                unsigned ldsm = (unsigned)(uintptr_t)&sM[r * LDSW + cc * 4];
                async_ld16(ldsm, (const void*)(M + (size_t)(row0 + r) * DIN + cc * 4));
            }
        }
    }

    // ---- B fragments: load once per wave, keep in registers across subtiles
    const int m     = lane & 15;
    const int khalf = (lane >> 4) << 1;        // 0 | 2
    const int ncol  = col_tile * 16 + m;

    v2f b1[DIN / 4];
    v2f b2[HASN ? DIN / 4 : 1];
    #pragma unroll
    for (int kk = 0; kk < DIN / 4; ++kk) {
        const int ka = kk * 4 + khalf;
        b1[kk].x = W1[ka * DOUT + ncol];
        b1[kk].y = W1[(ka + 1) * DOUT + ncol];
        if (HASN) {
            b2[kk].x = W2[ka * DOUT + ncol];
            b2[kk].y = W2[(ka + 1) * DOUT + ncol];
        }
    }
    const float bv = bias[ncol];

    asm volatile("s_wait_asynccnt 0" ::: "memory");
    __syncthreads();

    // ---- sweep row subtiles: ds_load_b64 -> v_wmma_f32_16x16x4_f32
    for (int st = 0; st < RT; ++st) {
        const int lrow = st * 16 + m;
        v8f acc = {};
        #pragma unroll
        for (int kk = 0; kk < DIN / 4; ++kk) {
            const int ka = kk * 4 + khalf;
            v2f af = *(const v2f*)&sA[lrow * LDSW + ka];
            acc = __builtin_amdgcn_wmma_f32_16x16x4_f32(
                false, af, false, b1[kk], (short)0, acc, false, false);
        }
        if (HASN) {
            const float s = invdeg[row0 + lrow];
            #pragma unroll
            for (int kk = 0; kk < DIN / 4; ++kk) {
                const int ka = kk * 4 + khalf;
                v2f af = *(const v2f*)&sM[lrow * LDSW + ka];
                af.x *= s;
                af.y *= s;
                acc = __builtin_amdgcn_wmma_f32_16x16x4_f32(
                    false, af, false, b2[kk], (short)0, acc, false, false);
            }
        }
        // C/D layout: VGPR j -> rows j (lanes 0-15) / j+8 (lanes 16-31)
        const int rbase = row0 + st * 16 + (lane >> 4) * 8;
        #pragma unroll
        for (int j = 0; j < 8; ++j) {
            float v = acc[j] + bv;
            if (RELU) v = fmaxf(v, 0.0f);
            Out[(size_t)(rbase + j) * DOUT + ncol] = v;
        }
    }
}

// ---------------------------------------------------------------------------
extern "C" void kernel_launch(void* const* d_in, const int* in_sizes, int n_in,
                              void* d_out, int out_size, void* d_ws, size_t ws_size,
                              hipStream_t stream) {
    (void)in_sizes; (void)n_in; (void)out_size; (void)ws_size;

    const float* feat = (const float*)d_in[0];
    const int*   src  = (const int*)d_in[1];
    const int*   dst  = (const int*)d_in[2];
    const float* Ws[3] = { (const float*)d_in[3], (const float*)d_in[6], (const float*)d_in[9]  };
    const float* Wn[3] = { (const float*)d_in[4], (const float*)d_in[7], (const float*)d_in[10] };
    const float* bs[3] = { (const float*)d_in[5], (const float*)d_in[8], (const float*)d_in[11] };
    const float* Wfc  = (const float*)d_in[12];
    const float* bfc  = (const float*)d_in[13];
    float* out = (float*)d_out;

    // workspace layout (floats)
    float* wsf    = (float*)d_ws;
    float* invdeg = wsf;                       // N floats (padded region 2^17)
    const size_t NH = (size_t)NN * HH;
    float* msg = wsf + (1u << 17);             // [N, H]
    float* hA  = msg + NH;                     // [N, H]
    float* hB  = hA + NH;                      // [N, H]

    const int T = 256;

    // degrees -> inv_deg
    zero_f32<<<1024, T, 0, stream>>>(invdeg, (size_t)NN);
    count_deg<<<(EE + T - 1) / T, T, 0, stream>>>(dst, invdeg, EE);
    finalize_invdeg<<<(NN + T - 1) / T, T, 0, stream>>>(invdeg, NN);

    const int scatter_blocks = (EE * 32) / T;   // one wave per edge
    const dim3 gemm_block(128);
    const dim3 grid_h(NN / 80, HH / 64);        // 1250 x 2
    const dim3 grid_c(NN / 80, CC / 64);        // 1250 x 1

    // ---- layer 0: feat -> hA
    zero_f32<<<4096, T, 0, stream>>>(msg, NH);
    scatter_edges<<<scatter_blocks, T, 0, stream>>>(feat, src, dst, msg, EE);
    sage_gemm<HH, true, true><<<grid_h, gemm_block, 0, stream>>>(
        feat, msg, invdeg, Ws[0], Wn[0], bs[0], hA);

    // ---- layer 1: hA -> hB
    zero_f32<<<4096, T, 0, stream>>>(msg, NH);
    scatter_edges<<<scatter_blocks, T, 0, stream>>>(hA, src, dst, msg, EE);
    sage_gemm<HH, true, true><<<grid_h, gemm_block, 0, stream>>>(
        hA, msg, invdeg, Ws[1], Wn[1], bs[1], hB);

    // ---- layer 2: hB -> hA  (reference relu's the last layer too)
    zero_f32<<<4096, T, 0, stream>>>(msg, NH);
    scatter_edges<<<scatter_blocks, T, 0, stream>>>(hB, src, dst, msg, EE);
    sage_gemm<HH, true, true><<<grid_h, gemm_block, 0, stream>>>(
        hB, msg, invdeg, Ws[2], Wn[2], bs[2], hA);

    // ---- fc: hA -> out, no relu
    sage_gemm<CC, false, false><<<grid_c, gemm_block, 0, stream>>>(
        hA, nullptr, invdeg, Wfc, nullptr, bfc, out);
}